// SEGNN_32615981646158
// MI455X (gfx1250) — compile-verified
//
#include <hip/hip_runtime.h>
#include <math.h>

// ---------------------------------------------------------------------------
// SEGNN forward for MI455X (gfx1250, wave32). All dense Linear layers run on
// V_WMMA_F32_16X16X4_F32. Weights are repacked once into zero-padded,
// pair-interleaved buffers (unconditional b64 weight loads). Tensor-product
// A operands are generated on the fly from LDS; multi-N-tile and
// 3-component-fused GEMM helpers amortize A generation and B loads.
// ---------------------------------------------------------------------------

typedef float f32x2 __attribute__((ext_vector_type(2)));
typedef float f32x8 __attribute__((ext_vector_type(8)));

#define INV_SQRT3 0.57735026918962576451f
#define SQRT3     1.73205080756887729353f

constexpr int al4(int x)  { return (x + 3) & ~3; }
constexpr int al16(int x) { return (x + 15) & ~15; }

__device__ __forceinline__ float sigmoidf_(float x) { return 1.0f / (1.0f + __expf(-x)); }
__device__ __forceinline__ float geluf_(float x) {
  float x3 = x * x * x;
  return 0.5f * x * (1.0f + tanhf(0.7978845608028653559f * (x + 0.044715f * x3)));
}

__device__ __forceinline__ f32x8 wmma4(f32x2 a, f32x2 b, f32x8 c) {
  return __builtin_amdgcn_wmma_f32_16x16x4_f32(false, a, false, b, (short)0, c, false, false);
}

__device__ __forceinline__ f32x2 ldpair(const float* __restrict__ Bp, int Np, int ka, int ncol) {
  // Pair packing: Bp[((k>>1)*Np + n)*2 + (k&1)]; ka is even -> {B[ka][n], B[ka+1][n]}.
  return *(const f32x2*)(Bp + 2 * ((size_t)(ka >> 1) * Np + ncol));
}

// Single 16-wide N tile, double accumulator to break the D->C chain.
// genP(row,k) valid for k < split; genD(row,k) valid for split <= k < Kpad
// (LDS feature arrays are zero-padded so genD yields 0 for k >= true K).
template <typename FP, typename FD>
__device__ __forceinline__ f32x8 wmma_gemm2(FP genP, FD genD, const int split,
                                            const int Kpad,
                                            const float* __restrict__ Bp,
                                            const int Np, const int n0) {
  const int lane = threadIdx.x & 31;
  const int hlf = lane >> 4, low = lane & 15;
  const int ncol = n0 + low;
  f32x8 acc0 = {}, acc1 = {};
  const int sLo = split & ~3;
  int k0 = 0;
  for (; k0 + 8 <= sLo; k0 += 8) {
    const int ka = k0 + 2 * hlf;
    f32x2 a0; a0.x = genP(low, ka);     a0.y = genP(low, ka + 1);
    f32x2 b0 = ldpair(Bp, Np, ka, ncol);
    const int kb = ka + 4;
    f32x2 a1; a1.x = genP(low, kb);     a1.y = genP(low, kb + 1);
    f32x2 b1 = ldpair(Bp, Np, kb, ncol);
    acc0 = wmma4(a0, b0, acc0);
    acc1 = wmma4(a1, b1, acc1);
  }
  if (k0 < sLo) {
    const int ka = k0 + 2 * hlf;
    f32x2 a; a.x = genP(low, ka); a.y = genP(low, ka + 1);
    acc0 = wmma4(a, ldpair(Bp, Np, ka, ncol), acc0);
    k0 += 4;
  }
  if (sLo < split) {  // single 4-chunk straddling the product/dot boundary
    const int ka = k0 + 2 * hlf;
    const bool p0 = ka < split, p1 = ka + 1 < split;
    const int kp0 = p0 ? ka : split - 1, kd0 = p0 ? split : ka;
    const int kp1 = p1 ? ka + 1 : split - 1, kd1 = p1 ? split : ka + 1;
    f32x2 a;
    a.x = p0 ? genP(low, kp0) : genD(low, kd0);
    a.y = p1 ? genP(low, kp1) : genD(low, kd1);
    acc0 = wmma4(a, ldpair(Bp, Np, ka, ncol), acc0);
    k0 += 4;
  }
  for (; k0 + 8 <= Kpad; k0 += 8) {
    const int ka = k0 + 2 * hlf;
    f32x2 a0; a0.x = genD(low, ka);     a0.y = genD(low, ka + 1);
    f32x2 b0 = ldpair(Bp, Np, ka, ncol);
    const int kb = ka + 4;
    f32x2 a1; a1.x = genD(low, kb);     a1.y = genD(low, kb + 1);
    f32x2 b1 = ldpair(Bp, Np, kb, ncol);
    acc0 = wmma4(a0, b0, acc0);
    acc1 = wmma4(a1, b1, acc1);
  }
  if (k0 < Kpad) {
    const int ka = k0 + 2 * hlf;
    f32x2 a; a.x = genD(low, ka); a.y = genD(low, ka + 1);
    acc0 = wmma4(a, ldpair(Bp, Np, ka, ncol), acc0);
  }
  return acc0 + acc1;
}

// NT adjacent 16-wide N tiles sharing one A generation per K-chunk.
// NT>=2 gives NT independent accumulator chains; NT==1 falls back to gemm2.
template <int NT, typename FP, typename FD>
__device__ __forceinline__ void wmma_gemm_n(FP genP, FD genD, const int split,
                                            const int Kpad,
                                            const float* __restrict__ Bp,
                                            const int Np, f32x8 (&acc)[NT]) {
  if constexpr (NT == 1) {
    acc[0] = wmma_gemm2(genP, genD, split, Kpad, Bp, Np, 0);
  } else {
    const int lane = threadIdx.x & 31;
    const int hlf = lane >> 4, low = lane & 15;
    const int sLo = split & ~3;
    int k0 = 0;
    for (; k0 < sLo; k0 += 4) {
      const int ka = k0 + 2 * hlf;
      f32x2 a; a.x = genP(low, ka); a.y = genP(low, ka + 1);
#pragma unroll
      for (int nt = 0; nt < NT; ++nt)
        acc[nt] = wmma4(a, ldpair(Bp, Np, ka, nt * 16 + low), acc[nt]);
    }
    if (sLo < split) {
      const int ka = k0 + 2 * hlf;
      const bool p0 = ka < split, p1 = ka + 1 < split;
      const int kp0 = p0 ? ka : split - 1, kd0 = p0 ? split : ka;
      const int kp1 = p1 ? ka + 1 : split - 1, kd1 = p1 ? split : ka + 1;
      f32x2 a;
      a.x = p0 ? genP(low, kp0) : genD(low, kd0);
      a.y = p1 ? genP(low, kp1) : genD(low, kd1);
#pragma unroll
      for (int nt = 0; nt < NT; ++nt)
        acc[nt] = wmma4(a, ldpair(Bp, Np, ka, nt * 16 + low), acc[nt]);
      k0 += 4;
    }
    for (; k0 < Kpad; k0 += 4) {
      const int ka = k0 + 2 * hlf;
      f32x2 a; a.x = genD(low, ka); a.y = genD(low, ka + 1);
#pragma unroll
      for (int nt = 0; nt < NT; ++nt)
        acc[nt] = wmma4(a, ldpair(Bp, Np, ka, nt * 16 + low), acc[nt]);
    }
  }
}

// Three component GEMMs (x,y,z of a vector path) sharing one B load per
// K-chunk; genP/genD take (row,k,comp). Three independent accumulator chains.
template <typename FP, typename FD>
__device__ __forceinline__ void wmma_gemm_vec3(FP genP, FD genD, const int split,
                                               const int Kpad,
                                               const float* __restrict__ Bp,
                                               const int Np, f32x8 (&acc)[3]) {
  const int lane = threadIdx.x & 31;
  const int hlf = lane >> 4, low = lane & 15;
  const int sLo = split & ~3;
  int k0 = 0;
  for (; k0 < sLo; k0 += 4) {
    const int ka = k0 + 2 * hlf;
    f32x2 b = ldpair(Bp, Np, ka, low);
#pragma unroll
    for (int c = 0; c < 3; ++c) {
      f32x2 a; a.x = genP(low, ka, c); a.y = genP(low, ka + 1, c);
      acc[c] = wmma4(a, b, acc[c]);
    }
  }
  if (sLo < split) {
    const int ka = k0 + 2 * hlf;
    const bool p0 = ka < split, p1 = ka + 1 < split;
    const int kp0 = p0 ? ka : split - 1, kd0 = p0 ? split : ka;
    const int kp1 = p1 ? ka + 1 : split - 1, kd1 = p1 ? split : ka + 1;
    f32x2 b = ldpair(Bp, Np, ka, low);
#pragma unroll
    for (int c = 0; c < 3; ++c) {
      f32x2 a;
      a.x = p0 ? genP(low, kp0, c) : genD(low, kd0, c);
      a.y = p1 ? genP(low, kp1, c) : genD(low, kd1, c);
      acc[c] = wmma4(a, b, acc[c]);
    }
    k0 += 4;
  }
  for (; k0 < Kpad; k0 += 4) {
    const int ka = k0 + 2 * hlf;
    f32x2 b = ldpair(Bp, Np, ka, low);
#pragma unroll
    for (int c = 0; c < 3; ++c) {
      f32x2 a; a.x = genD(low, ka, c); a.y = genD(low, ka + 1, c);
      acc[c] = wmma4(a, b, acc[c]);
    }
  }
}

// ------------------------------- utility kernels ---------------------------

__global__ void zero_kernel(float* __restrict__ p, long n) {
  for (long i = (long)blockIdx.x * blockDim.x + threadIdx.x; i < n;
       i += (long)gridDim.x * blockDim.x)
    p[i] = 0.0f;
}

// Repack weight [K x N] row-major into zero-padded pair-interleaved
// [Kp/2 x Np x 2] so each even k yields an 8B-aligned {B[k][n],B[k+1][n]} pair.
__global__ void pack_pairs_kernel(const float* __restrict__ src, int K, int N,
                                  float* __restrict__ dst, int Np, long total) {
  for (long i = (long)blockIdx.x * blockDim.x + threadIdx.x; i < total;
       i += (long)gridDim.x * blockDim.x) {
    long pair = i >> 1;
    int n = (int)(pair % Np);
    int k = (int)(pair / Np) * 2 + (int)(i & 1);
    dst[i] = (k < K && n < N) ? src[(size_t)k * N + n] : 0.0f;
  }
}

__global__ void init_kernel(const float* __restrict__ pos, const float* __restrict__ vel,
                            const float* __restrict__ charge,
                            float* __restrict__ sA, float* __restrict__ vA, long Nn) {
  long n = (long)blockIdx.x * blockDim.x + threadIdx.x;
  if (n >= Nn) return;
  sA[n * 32] = charge[n];
  vA[n * 30 + 0] = pos[n * 3 + 0];
  vA[n * 30 + 1] = pos[n * 3 + 1];
  vA[n * 30 + 2] = pos[n * 3 + 2];
  vA[n * 30 + 3] = vel[n * 3 + 0];
  vA[n * 30 + 4] = vel[n * 3 + 1];
  vA[n * 30 + 5] = vel[n * 3 + 2];
}

__global__ void edge_pre_kernel(const float* __restrict__ pos,
                                const int* __restrict__ snd, const int* __restrict__ rcv,
                                float* __restrict__ av, float* __restrict__ sumAV,
                                float* __restrict__ cnt, long Ee) {
  long e = (long)blockIdx.x * blockDim.x + threadIdx.x;
  if (e >= Ee) return;
  int s = snd[e], r = rcv[e];
  float dx = pos[(size_t)s * 3 + 0] - pos[(size_t)r * 3 + 0];
  float dy = pos[(size_t)s * 3 + 1] - pos[(size_t)r * 3 + 1];
  float dz = pos[(size_t)s * 3 + 2] - pos[(size_t)r * 3 + 2];
  float k = SQRT3 / (sqrtf(dx * dx + dy * dy + dz * dz) + 1e-9f);
  float ax = dx * k, ay = dy * k, az = dz * k;
  av[e * 3 + 0] = ax; av[e * 3 + 1] = ay; av[e * 3 + 2] = az;
  atomicAdd(&sumAV[(size_t)r * 3 + 0], ax);
  atomicAdd(&sumAV[(size_t)r * 3 + 1], ay);
  atomicAdd(&sumAV[(size_t)r * 3 + 2], az);
  atomicAdd(&cnt[r], 1.0f);
}

__global__ void assemble_kernel(const float* __restrict__ sA, const float* __restrict__ vA,
                                float* __restrict__ out, long Nn) {
  long n = (long)blockIdx.x * blockDim.x + threadIdx.x;
  if (n >= Nn) return;
  out[n * 7] = sA[n * 32];
  for (int t = 0; t < 6; ++t) out[n * 7 + 1 + t] = vA[n * 30 + t];
}

// ------------------------------- edge MLP ----------------------------------
template <int STEP>
__global__ __launch_bounds__(128) void edge_mlp_kernel(
    const float* __restrict__ sIn, const float* __restrict__ vIn,
    const float* __restrict__ av,
    const int* __restrict__ senders, const int* __restrict__ receivers,
    const float* __restrict__ e0_ws, const float* __restrict__ e0_bs,
    const float* __restrict__ e0_wv,
    const float* __restrict__ e1_ws, const float* __restrict__ e1_bs,
    const float* __restrict__ e1_wv,
    float* __restrict__ aggS, float* __restrict__ aggV, long Ee) {
  constexpr int D0  = STEP ? 32 : 1;
  constexpr int D1  = STEP ? 10 : 2;
  constexpr int XSN = 2 * D0;          // concat scalars (= product count)
  constexpr int XVN = 2 * D1;          // concat vectors (= dot count)
  constexpr int MVP = XVN + 2;         // +2 zero pad vectors for K padding
  constexpr int K0P = al4(XSN + XVN);
  constexpr int HVP = 12;              // 10 + 2 zero pad vectors
  constexpr int PW  = 16 * (3 + XSN + 3 * MVP + 42 + 3 * HVP);
  __shared__ float smem[4 * PW];
  __shared__ int   rsh[4][16];
  const int wave = threadIdx.x >> 5, lane = threadIdx.x & 31;
  float* W  = smem + wave * PW;
  float* AV = W;                          // 16 x 3
  float* MS = AV + 16 * 3;                // 16 x XSN
  float* MV = MS + 16 * XSN;              // 16 x MVP x 3 (zero-padded)
  float* HS = MV + 16 * MVP * 3;          // 16 x 42
  float* HV = HS + 16 * 42;               // 16 x HVP x 3 (zero-padded)
  const long ebase = ((long)blockIdx.x * 4 + wave) * 16;

  __builtin_prefetch(e0_ws, 0, 1);
  __builtin_prefetch(e1_ws, 0, 1);

  // ---- stage ----
  for (int i = lane; i < 16; i += 32) {
    long e = ebase + i;
    rsh[wave][i] = (e < Ee) ? receivers[e] : 0;
  }
  for (int idx = lane; idx < 16 * 3; idx += 32) {
    int r = idx / 3, c = idx % 3;
    long e = ebase + r;
    AV[idx] = (e < Ee) ? av[e * 3 + c] : 0.0f;
  }
  for (int idx = lane; idx < 16 * XSN; idx += 32) {
    int r = idx / XSN, i = idx % XSN;
    long e = ebase + r;
    float val = 0.0f;
    if (e < Ee) {
      int node = (i < D0) ? senders[e] : receivers[e];
      int ii   = (i < D0) ? i : i - D0;
      val = sIn[(size_t)node * 32 + ii];
    }
    MS[idx] = val;
  }
  for (int idx = lane; idx < 16 * MVP * 3; idx += 32) {
    int r = idx / (MVP * 3), t = idx % (MVP * 3), i = t / 3, c = t % 3;
    long e = ebase + r;
    float val = 0.0f;
    if (e < Ee && i < XVN) {
      int node = (i < D1) ? senders[e] : receivers[e];
      int ii   = (i < D1) ? i : i - D1;
      val = vIn[(size_t)node * 30 + ii * 3 + c];
    }
    MV[idx] = val;
  }
  for (int idx = lane; idx < 16 * 6; idx += 32) {  // zero HV pad vectors 10,11
    int r = idx / 6, t = idx % 6;
    HV[r * (3 * HVP) + 30 + t] = 0.0f;
  }
  __syncthreads();

  const int hlf = lane >> 4, low = lane & 15;

  // ---- tplg0 scalar (K0 -> 42), split at XSN, 3 fused N tiles ----
  auto ts0P = [&](int r, int k) -> float { return MS[r * XSN + k]; };
  auto ts0D = [&](int r, int k) -> float {
    int i = k - XSN;
    const float* mv = &MV[(r * MVP + i) * 3];
    const float* a3 = &AV[r * 3];
    return (mv[0] * a3[0] + mv[1] * a3[1] + mv[2] * a3[2]) * INV_SQRT3;
  };
  {
    f32x8 accs[3] = {};
    wmma_gemm_n<3>(ts0P, ts0D, XSN, K0P, e0_ws, 48, accs);
    for (int nt = 0; nt < 3; ++nt) {
      int col = nt * 16 + low;
      if (col < 42) {
        float bias = e0_bs[col];
        for (int j = 0; j < 8; ++j) HS[(j + 8 * hlf) * 42 + col] = accs[nt][j] + bias;
      }
    }
  }
  // ---- tplg0 vector (K0 -> 10, 3 components fused) ----
  {
    auto vP = [&](int r, int k, int c) -> float { return MS[r * XSN + k] * AV[r * 3 + c]; };
    auto vD = [&](int r, int k, int c) -> float { return MV[(r * MVP + (k - XSN)) * 3 + c]; };
    f32x8 acc[3] = {};
    wmma_gemm_vec3(vP, vD, XSN, K0P, e0_wv, 16, acc);
    if (low < 10)
      for (int c = 0; c < 3; ++c)
        for (int j = 0; j < 8; ++j) HV[(j + 8 * hlf) * (3 * HVP) + low * 3 + c] = acc[c][j];
  }
  __syncthreads();

  // ---- gate ----
  for (int idx = lane; idx < 16 * 30; idx += 32) {
    int r = idx / 30, oc = idx % 30, o = oc / 3;
    HV[r * (3 * HVP) + oc] *= sigmoidf_(HS[r * 42 + o]);
  }
  for (int idx = lane; idx < 16 * 32; idx += 32) {
    int r = idx / 32, o = idx % 32;
    HS[r * 42 + 10 + o] = geluf_(HS[r * 42 + 10 + o]);
  }
  __syncthreads();

  // ---- tplg1 scalar (42 -> 32), split at 32, 2 fused N tiles, scatter ----
  auto ts1P = [&](int r, int k) -> float { return HS[r * 42 + 10 + k]; };
  auto ts1D = [&](int r, int k) -> float {
    int i = k - 32;
    const float* hv = &HV[(r * HVP + i) * 3];
    const float* a3 = &AV[r * 3];
    return (hv[0] * a3[0] + hv[1] * a3[1] + hv[2] * a3[2]) * INV_SQRT3;
  };
  {
    f32x8 accs[2] = {};
    wmma_gemm_n<2>(ts1P, ts1D, 32, 44, e1_ws, 32, accs);
    for (int nt = 0; nt < 2; ++nt) {
      int col = nt * 16 + low;  // < 32
      float bias = e1_bs[col];
      for (int j = 0; j < 8; ++j) {
        int r = j + 8 * hlf;
        long e = ebase + r;
        if (e < Ee) atomicAdd(&aggS[(size_t)rsh[wave][r] * 32 + col], accs[nt][j] + bias);
      }
    }
  }
  // ---- tplg1 vector (42 -> 10, 3 components fused), scatter ----
  {
    auto vP = [&](int r, int k, int c) -> float { return HS[r * 42 + 10 + k] * AV[r * 3 + c]; };
    auto vD = [&](int r, int k, int c) -> float { return HV[(r * HVP + (k - 32)) * 3 + c]; };
    f32x8 acc[3] = {};
    wmma_gemm_vec3(vP, vD, 32, 44, e1_wv, 16, acc);
    if (low < 10)
      for (int c = 0; c < 3; ++c)
        for (int j = 0; j < 8; ++j) {
          int r = j + 8 * hlf;
          long e = ebase + r;
          if (e < Ee) atomicAdd(&aggV[(size_t)rsh[wave][r] * 30 + low * 3 + c], acc[c][j]);
        }
  }
}

// ------------------------------- node MLP ----------------------------------
template <int STEP>
__global__ __launch_bounds__(128) void node_mlp_kernel(
    const float* __restrict__ sIn, const float* __restrict__ vIn,
    const float* __restrict__ aggS, const float* __restrict__ aggV,
    const float* __restrict__ sumAV, const float* __restrict__ cnt,
    const float* __restrict__ n0_ws, const float* __restrict__ n0_bs,
    const float* __restrict__ n0_wv,
    const float* __restrict__ n1_ws, const float* __restrict__ n1_bs,
    const float* __restrict__ n1_wv,
    const float* __restrict__ r_ws, const float* __restrict__ r_bs,
    const float* __restrict__ r_wv,
    float* __restrict__ sOut, float* __restrict__ vOut, long Nn) {
  constexpr int D0 = STEP ? 32 : 1;
  constexpr int D1 = STEP ? 10 : 2;
  constexpr int H0 = STEP ? 1 : 32;
  constexpr int H1 = STEP ? 2 : 10;
  constexpr int D1P = D1 + 1;          // +1 zero pad vector
  constexpr int H1P = H1 + 1;          // +1 zero pad vector
  constexpr int KA0P = al4(D0 * 33 + D1 * 11);
  constexpr int KB0P = al4(D0 * 11 + D1 * 33);
  constexpr int KA1P = al4(H0 * 33 + H1 * 11);
  constexpr int KB1P = al4(H0 * 11 + H1 * 33);
  constexpr int HSW = H0 + H1;
  constexpr int NP0 = al16(HSW);
  constexpr int NP1 = al16(H0);
  constexpr int NT0 = NP0 / 16;
  constexpr int NT1 = NP1 / 16;
  constexpr int PW  = 16 * (D0 + 3 * D1P + 33 + 33 + HSW + 3 * H1P);
  __shared__ float smem[4 * PW];
  const int wave = threadIdx.x >> 5, lane = threadIdx.x & 31;
  float* W   = smem + wave * PW;
  float* XSL = W;                   // 16 x D0
  float* XVL = XSL + 16 * D0;       // 16 x D1P x 3 (zero-padded)
  float* YS  = XVL + 16 * D1P * 3;  // 16 x 33
  float* YV  = YS + 16 * 33;        // 16 x 11 x 3
  float* HS  = YV + 16 * 33;        // 16 x HSW
  float* HV  = HS + 16 * HSW;       // 16 x H1P x 3 (zero-padded)
  const long nbase = ((long)blockIdx.x * 4 + wave) * 16;

  __builtin_prefetch(n1_ws, 0, 1);

  // ---- stage ----
  for (int idx = lane; idx < 16 * D0; idx += 32) {
    int r = idx / D0, i = idx % D0;
    long n = nbase + r;
    XSL[idx] = (n < Nn) ? sIn[(size_t)n * 32 + i] : 0.0f;
  }
  for (int idx = lane; idx < 16 * D1P * 3; idx += 32) {
    int r = idx / (D1P * 3), t = idx % (D1P * 3), i = t / 3, c = t % 3;
    long n = nbase + r;
    XVL[idx] = (n < Nn && i < D1) ? vIn[(size_t)n * 30 + i * 3 + c] : 0.0f;
  }
  for (int idx = lane; idx < 16 * 33; idx += 32) {
    int r = idx / 33, j = idx % 33;
    long n = nbase + r;
    float val = 0.0f;
    if (n < Nn) {
      float c = cnt[n];
      float invc = 1.0f / fmaxf(c, 1.0f);
      val = (j < 32) ? aggS[(size_t)n * 32 + j] * invc : (c > 0.0f ? 1.0f : 0.0f);
    }
    YS[idx] = val;
  }
  for (int idx = lane; idx < 16 * 33; idx += 32) {
    int r = idx / 33, t = idx % 33;
    long n = nbase + r;
    float val = 0.0f;
    if (n < Nn) {
      float c = cnt[n];
      float invc = 1.0f / fmaxf(c, 1.0f);
      val = (t < 30) ? aggV[(size_t)n * 30 + t] * invc : sumAV[(size_t)n * 3 + (t - 30)] * invc;
    }
    YV[idx] = val;
  }
  for (int idx = lane; idx < 16 * 3; idx += 32) {  // zero HV pad vector i=H1
    int r = idx / 3, c = idx % 3;
    HV[r * (3 * H1P) + H1 * 3 + c] = 0.0f;
  }
  __syncthreads();

  const int hlf = lane >> 4, low = lane & 15;

  // ---- tplg0 scalar (KA0 -> HSW), split at D0*33 ----
  auto tsA0P = [&](int r, int k) -> float {
    int i = k / 33, j = k - i * 33;
    return XSL[r * D0 + i] * YS[r * 33 + j];
  };
  auto tsA0D = [&](int r, int k) -> float {
    int t = k - D0 * 33, i = t / 11, j = t - i * 11;
    const float* xv = &XVL[(r * D1P + i) * 3];
    const float* yv = &YV[r * 33 + j * 3];
    return (xv[0] * yv[0] + xv[1] * yv[1] + xv[2] * yv[2]) * INV_SQRT3;
  };
  {
    f32x8 accs[NT0] = {};
    wmma_gemm_n<NT0>(tsA0P, tsA0D, D0 * 33, KA0P, n0_ws, NP0, accs);
    for (int nt = 0; nt < NT0; ++nt) {
      int col = nt * 16 + low;
      if (col < HSW) {
        float b = n0_bs[col];
        for (int j = 0; j < 8; ++j) HS[(j + 8 * hlf) * HSW + col] = accs[nt][j] + b;
      }
    }
  }
  // ---- tplg0 vector (KB0 -> H1, 3 components fused), split at D0*11 ----
  {
    auto vP = [&](int r, int k, int c) -> float {
      int i = k / 11, j = k - i * 11;
      return XSL[r * D0 + i] * YV[r * 33 + j * 3 + c];
    };
    auto vD = [&](int r, int k, int c) -> float {
      int t = k - D0 * 11, i = t / 33, j = t - i * 33;
      return XVL[(r * D1P + i) * 3 + c] * YS[r * 33 + j];
    };
    f32x8 acc[3] = {};
    wmma_gemm_vec3(vP, vD, D0 * 11, KB0P, n0_wv, 16, acc);
    if (low < H1)
      for (int c = 0; c < 3; ++c)
        for (int j = 0; j < 8; ++j) HV[(j + 8 * hlf) * (3 * H1P) + low * 3 + c] = acc[c][j];
  }
  __syncthreads();

  // ---- gate ----
  for (int idx = lane; idx < 16 * 3 * H1; idx += 32) {
    int r = idx / (3 * H1), oc = idx % (3 * H1), o = oc / 3;
    HV[r * (3 * H1P) + oc] *= sigmoidf_(HS[r * HSW + o]);
  }
  for (int idx = lane; idx < 16 * H0; idx += 32) {
    int r = idx / H0, o = idx % H0;
    HS[r * HSW + H1 + o] = geluf_(HS[r * HSW + H1 + o]);
  }
  __syncthreads();

  // ---- tplg1 scalar (KA1 -> H0) + bias + residual, split at H0*33 ----
  auto tsA1P = [&](int r, int k) -> float {
    int i = k / 33, j = k - i * 33;
    return HS[r * HSW + H1 + i] * YS[r * 33 + j];
  };
  auto tsA1D = [&](int r, int k) -> float {
    int t = k - H0 * 33, i = t / 11, j = t - i * 11;
    const float* hv = &HV[(r * H1P + i) * 3];
    const float* yv = &YV[r * 33 + j * 3];
    return (hv[0] * yv[0] + hv[1] * yv[1] + hv[2] * yv[2]) * INV_SQRT3;
  };
  {
    f32x8 accs[NT1] = {};
    wmma_gemm_n<NT1>(tsA1P, tsA1D, H0 * 33, KA1P, n1_ws, NP1, accs);
    for (int nt = 0; nt < NT1; ++nt) {
      int col = nt * 16 + low;
      if (col < H0) {
        float b = n1_bs[col];
        for (int j = 0; j < 8; ++j) {
          int r = j + 8 * hlf;
          long n = nbase + r;
          if (n < Nn) {
            float res = r_bs[col];
            for (int i = 0; i < D0; ++i) res += XSL[r * D0 + i] * r_ws[i * H0 + col];
            sOut[(size_t)n * 32 + col] = accs[nt][j] + b + res;
          }
        }
      }
    }
  }
  // ---- tplg1 vector (KB1 -> H1, 3 components fused) + residual ----
  {
    auto vP = [&](int r, int k, int c) -> float {
      int i = k / 11, j = k - i * 11;
      return HS[r * HSW + H1 + i] * YV[r * 33 + j * 3 + c];
    };
    auto vD = [&](int r, int k, int c) -> float {
      int t = k - H0 * 11, i = t / 33, j = t - i * 33;
      return HV[(r * H1P + i) * 3 + c] * YS[r * 33 + j];
    };
    f32x8 acc[3] = {};
    wmma_gemm_vec3(vP, vD, H0 * 11, KB1P, n1_wv, 16, acc);
    if (low < H1)
      for (int c = 0; c < 3; ++c)
        for (int j = 0; j < 8; ++j) {
          int r = j + 8 * hlf;
          long n = nbase + r;
          if (n < Nn) {
            float res = 0.0f;
            for (int i = 0; i < D1; ++i) res += XVL[(r * D1P + i) * 3 + c] * r_wv[i * H1 + low];
            vOut[(size_t)n * 30 + low * 3 + c] = acc[c][j] + res;
          }
        }
  }
}

// ------------------------------- launch ------------------------------------
// Param leaf order (jax tree flatten: dict keys sorted, lists in order):
// per step s (base = s*15): +0 edge0.bs +1 edge0.ws +2 edge0.wv
//                           +3 edge1.bs +4 edge1.ws +5 edge1.wv
//                           +6 node0.bs +7 node0.ws +8 node0.wv
//                           +9 node1.bs +10 node1.ws +11 node1.wv
//                           +12 res.bs  +13 res.ws  +14 res.wv
// Then: 30=pos, 31=vel, 32=charge, 33=senders, 34=receivers.
extern "C" void kernel_launch(void* const* d_in, const int* in_sizes, int n_in,
                              void* d_out, int out_size, void* d_ws, size_t ws_size,
                              hipStream_t stream) {
  auto F = [&](int i) { return (const float*)d_in[i]; };
  const float* pos    = F(30);
  const float* vel    = F(31);
  const float* charge = F(32);
  const int* snd = (const int*)d_in[33];
  const int* rcv = (const int*)d_in[34];
  const long Nn = in_sizes[30] / 3;
  const long Ee = in_sizes[33];

  // Workspace layout (floats): 190*N + 3*E + ~95k padded weights (~45 MB).
  float* ws    = (float*)d_ws;
  float* sA    = ws;              // N x 32
  float* vA    = sA + 32 * Nn;    // N x 30
  float* sB    = vA + 30 * Nn;    // N x 32
  float* vB    = sB + 32 * Nn;    // N x 30
  float* aggS  = vB + 30 * Nn;    // N x 32
  float* aggV  = aggS + 32 * Nn;  // N x 30
  float* sumAV = aggV + 30 * Nn;  // N x 3
  float* cnt   = sumAV + 3 * Nn;  // N
  float* av    = cnt + Nn;        // E x 3
  float* cur   = av + 3 * Ee;     // padded weights start here

  const int T = 256;
  auto pack = [&](const float* src, int K, int N, int Kp, int Np) -> float* {
    float* dst = cur;
    long tot = (long)Kp * Np;
    cur += tot;
    pack_pairs_kernel<<<(unsigned)((tot + T - 1) / T), T, 0, stream>>>(src, K, N, dst, Np, tot);
    return dst;
  };
  // step 0 padded weights
  float* p0_e0_ws = pack(F(1),  6,    42, 8,    48);
  float* p0_e0_wv = pack(F(2),  6,    10, 8,    16);
  float* p0_e1_ws = pack(F(4),  42,   32, 44,   32);
  float* p0_e1_wv = pack(F(5),  42,   10, 44,   16);
  float* p0_n0_ws = pack(F(7),  55,   42, 56,   48);
  float* p0_n0_wv = pack(F(8),  77,   10, 80,   16);
  float* p0_n1_ws = pack(F(10), 1166, 32, 1168, 32);
  float* p0_n1_wv = pack(F(11), 682,  10, 684,  16);
  // step 1 padded weights
  float* p1_e0_ws = pack(F(16), 84,   42, 84,   48);
  float* p1_e0_wv = pack(F(17), 84,   10, 84,   16);
  float* p1_e1_ws = pack(F(19), 42,   32, 44,   32);
  float* p1_e1_wv = pack(F(20), 42,   10, 44,   16);
  float* p1_n0_ws = pack(F(22), 1166, 3,  1168, 16);
  float* p1_n0_wv = pack(F(23), 682,  2,  684,  16);
  float* p1_n1_ws = pack(F(25), 55,   1,  56,   16);
  float* p1_n1_wv = pack(F(26), 77,   2,  80,   16);

  const unsigned eb = (unsigned)(((Ee + 15) / 16 + 3) / 4);
  const unsigned nb = (unsigned)(((Nn + 15) / 16 + 3) / 4);

  zero_kernel<<<2048, T, 0, stream>>>(sumAV, 4 * Nn);  // sumAV + cnt
  init_kernel<<<(unsigned)((Nn + T - 1) / T), T, 0, stream>>>(pos, vel, charge, sA, vA, Nn);
  edge_pre_kernel<<<(unsigned)((Ee + T - 1) / T), T, 0, stream>>>(pos, snd, rcv, av, sumAV, cnt, Ee);

  // ---- step 0: (1,2) -> (32,10) ----
  zero_kernel<<<2048, T, 0, stream>>>(aggS, 62 * Nn);
  edge_mlp_kernel<0><<<eb, 128, 0, stream>>>(
      sA, vA, av, snd, rcv,
      p0_e0_ws, F(0), p0_e0_wv,
      p0_e1_ws, F(3), p0_e1_wv,
      aggS, aggV, Ee);
  node_mlp_kernel<0><<<nb, 128, 0, stream>>>(
      sA, vA, aggS, aggV, sumAV, cnt,
      p0_n0_ws, F(6), p0_n0_wv,
      p0_n1_ws, F(9), p0_n1_wv,
      F(13), F(12), F(14),
      sB, vB, Nn);

  // ---- step 1: (32,10) -> (1,2) ----
  zero_kernel<<<2048, T, 0, stream>>>(aggS, 62 * Nn);
  edge_mlp_kernel<1><<<eb, 128, 0, stream>>>(
      sB, vB, av, snd, rcv,
      p1_e0_ws, F(15), p1_e0_wv,
      p1_e1_ws, F(18), p1_e1_wv,
      aggS, aggV, Ee);
  node_mlp_kernel<1><<<nb, 128, 0, stream>>>(
      sB, vB, aggS, aggV, sumAV, cnt,
      p1_n0_ws, F(21), p1_n0_wv,
      p1_n1_ws, F(24), p1_n1_wv,
      F(28), F(27), F(29),
      sA, vA, Nn);

  assemble_kernel<<<(unsigned)((Nn + T - 1) / T), T, 0, stream>>>(sA, vA, (float*)d_out, Nn);
}